// ParticleFilterNetwork_1331439862214
// MI455X (gfx1250) — compile-verified
//
#include <hip/hip_runtime.h>

// ---------------------------------------------------------------------------
// Particle filter step, MI455X (gfx1250, wave32).
// Memory-bound (~650MB HBM traffic vs ~4.3 GFLOP => ~28us roofline @23.3TB/s).
// Fused predict+measurement pass using V_WMMA_F32_16X16X4_F32:
//   states_pred = states_prev @ (I + W_dyn) + ctrl@W_ctrl + 0.1*noise
//   (identity folded into the B fragments; control term folded into the
//    WMMA accumulator init since D columns are lane-indexed)
//   pred_obs    = states_pred @ W_obs   (re-fragmented via padded LDS tile)
// ---------------------------------------------------------------------------

#define N_B 64
#define M_P 16384
#define S_D 32
#define C_D 16
#define O_D 32
#define NEG_INF -3.402823466e38f

typedef __attribute__((ext_vector_type(2))) float v2f;
typedef __attribute__((ext_vector_type(8))) float v8f;

__device__ __forceinline__ v8f wmma4(v2f a, v2f b, v8f c) {
  // (neg_a, A, neg_b, B, c_mod, C, reuse_a, reuse_b)
  return __builtin_amdgcn_wmma_f32_16x16x4_f32(false, a, false, b, (short)0, c,
                                               false, false);
}

// ---------------------------------------------------------------------------
// Kernel A: fused predict + log-likelihood. One 16-particle tile per wave.
//   grid = (M/128, N), block = 256 (8 waves)
// ---------------------------------------------------------------------------
__global__ __launch_bounds__(256) void pf_predict_ll(
    const float* __restrict__ sp, const float* __restrict__ lwp,
    const float* __restrict__ obs, const float* __restrict__ ctrl,
    const float* __restrict__ Wd, const float* __restrict__ Wc,
    const float* __restrict__ Wo, const float* __restrict__ noise,
    float* __restrict__ pred, float* __restrict__ lwout) {
  // per-wave LDS staging tile (16 rows x 32 cols, stride 36 floats: 36*r+c
  // hits distinct banks for r=0..15, rows stay 8B-aligned for b64 loads)
  __shared__ float stage[8][16 * 36];

  const int n = blockIdx.y;
  const int w = threadIdx.x >> 5;
  const int lane = threadIdx.x & 31;
  const int lr = lane & 15;   // row (A) / col (B,C,D) within 16
  const int lh = lane >> 4;   // half select
  const int m0 = (blockIdx.x * 8 + w) * 16;
  const size_t baseNM = (size_t)n * M_P;

  // ---- B fragments: (I + W_dyn) and W_obs (contraction over first index) ----
  // B 4x16 f32 fragment: lane holds col t=lr, VGPR pair holds K = 4kc+2lh+{0,1}
  v2f bd[2][8], bo[2][8];
#pragma unroll
  for (int ct = 0; ct < 2; ++ct) {
#pragma unroll
    for (int kc = 0; kc < 8; ++kc) {
      const int k0 = 4 * kc + 2 * lh;
      const int t = 16 * ct + lr;
      v2f d, o;
      d.x = Wd[(k0 + 0) * S_D + t] + ((k0 + 0) == t ? 1.0f : 0.0f);
      d.y = Wd[(k0 + 1) * S_D + t] + ((k0 + 1) == t ? 1.0f : 0.0f);
      o.x = Wo[(k0 + 0) * O_D + t];
      o.y = Wo[(k0 + 1) * O_D + t];
      bd[ct][kc] = d;
      bo[ct][kc] = o;
    }
  }

  // ---- control term (controls[n] @ W_ctrl)[t]; D columns are lane-indexed,
  //      so it seeds every accumulator element directly ----
  v8f acc0, acc1;
#pragma unroll
  for (int ct = 0; ct < 2; ++ct) {
    const int t = 16 * ct + lr;
    float s = 0.f;
#pragma unroll
    for (int c = 0; c < C_D; ++c) s += ctrl[n * C_D + c] * Wc[c * S_D + t];
#pragma unroll
    for (int r = 0; r < 8; ++r) {
      if (ct == 0) acc0[r] = s; else acc1[r] = s;
    }
  }

  // ---- A fragments of states_prev tile: lane holds row m0+lr,
  //      K = 4kc+2lh+{0,1} (contiguous 8B per lane per fragment) ----
  const float* sprow = sp + (baseNM + m0 + lr) * S_D;
  v2f a[8];
#pragma unroll
  for (int kc = 0; kc < 8; ++kc)
    a[kc] = *(const v2f*)(sprow + 4 * kc + 2 * lh);

  // ---- GEMM 1: states_prev @ (I + W_dyn), seeded with control term ----
#pragma unroll
  for (int kc = 0; kc < 8; ++kc) {
    acc0 = wmma4(a[kc], bd[0][kc], acc0);
    acc1 = wmma4(a[kc], bd[1][kc], acc1);
  }

  // ---- elementwise in D layout: element r of v8f is (m0+r+8lh, 16ct+lr) ----
  float* stg = &stage[w][0];
#pragma unroll
  for (int r = 0; r < 8; ++r) {
    const int row = r + 8 * lh;
    const size_t rowb = (baseNM + m0 + row) * S_D;
    {
      const int t = lr;
      float v = acc0[r] + 0.1f * noise[rowb + t];
      pred[rowb + t] = v;
      stg[row * 36 + t] = v;
    }
    {
      const int t = 16 + lr;
      float v = acc1[r] + 0.1f * noise[rowb + t];
      pred[rowb + t] = v;
      stg[row * 36 + t] = v;
    }
  }
  __syncthreads();  // cross-lane LDS dependency before re-fragmenting

  // ---- re-fragment states_pred as A from LDS ----
  v2f a2[8];
#pragma unroll
  for (int kc = 0; kc < 8; ++kc)
    a2[kc] = *(const v2f*)(stg + lr * 36 + 4 * kc + 2 * lh);

  // ---- GEMM 2: states_pred @ W_obs ----
  v8f oa0 = {}, oa1 = {};
#pragma unroll
  for (int kc = 0; kc < 8; ++kc) {
    oa0 = wmma4(a2[kc], bo[0][kc], oa0);
    oa1 = wmma4(a2[kc], bo[1][kc], oa1);
  }

  // ---- log-likelihood: -0.5 * sum_o (pred_obs - obs)^2, row reduction ----
  const float ob0 = obs[n * O_D + lr];
  const float ob1 = obs[n * O_D + 16 + lr];
#pragma unroll
  for (int r = 0; r < 8; ++r) {
    float e0 = oa0[r] - ob0;
    float e1 = oa1[r] - ob1;
    float ss = e0 * e0 + e1 * e1;
    ss += __shfl_xor(ss, 1);
    ss += __shfl_xor(ss, 2);
    ss += __shfl_xor(ss, 4);
    ss += __shfl_xor(ss, 8);  // masks < 16 stay within each 16-lane half
    if (lr == 0) {
      const int m = m0 + r + 8 * lh;
      lwout[baseNM + m] = lwp[baseNM + m] - 0.5f * ss;
    }
  }
}

// ---------------------------------------------------------------------------
// Kernel B: per-batch logsumexp over M.  grid = N, block = 256
// ---------------------------------------------------------------------------
__global__ __launch_bounds__(256) void pf_lse(const float* __restrict__ lw,
                                              float* __restrict__ Z) {
  __shared__ float red[256];
  const int n = blockIdx.x, t = threadIdx.x;
  const float* row = lw + (size_t)n * M_P;
  float mx = NEG_INF;
  for (int i = t; i < M_P; i += 256) mx = fmaxf(mx, row[i]);
  red[t] = mx;
  __syncthreads();
  for (int s = 128; s > 0; s >>= 1) {
    if (t < s) red[t] = fmaxf(red[t], red[t + s]);
    __syncthreads();
  }
  const float gm = red[0];
  __syncthreads();
  float sm = 0.f;
  for (int i = t; i < M_P; i += 256) sm += __expf(row[i] - gm);
  red[t] = sm;
  __syncthreads();
  for (int s = 128; s > 0; s >>= 1) {
    if (t < s) red[t] += red[t + s];
    __syncthreads();
  }
  if (t == 0) Z[n] = gm + __logf(red[0]);
}

__global__ void pf_zero(float* __restrict__ p, int n) {
  const int i = blockIdx.x * blockDim.x + threadIdx.x;
  if (i < n) p[i] = 0.f;
}

// ---------------------------------------------------------------------------
// Kernel C: weighted state estimates.  grid = (8, N), block = 256
//   lane = state dim (coalesced 128B rows), group = particle stripe.
// ---------------------------------------------------------------------------
__global__ __launch_bounds__(256) void pf_estimate(
    const float* __restrict__ pred, const float* __restrict__ lw,
    const float* __restrict__ Z, float* __restrict__ est) {
  __shared__ float red[256];
  const int n = blockIdx.y;
  const int s = threadIdx.x & 31, g = threadIdx.x >> 5;
  const float z = Z[n];
  const size_t base = (size_t)n * M_P;
  const int chunk = M_P / 8;  // gridDim.x == 8
  const int p0 = blockIdx.x * chunk;
  float acc = 0.f;
  for (int i = g; i < chunk; i += 8) {
    const int m = p0 + i;
    const float wgt = __expf(lw[base + m] - z);
    acc += wgt * pred[(base + m) * S_D + s];
  }
  red[threadIdx.x] = acc;
  __syncthreads();
  if (g == 0) {
    float t = 0.f;
#pragma unroll
    for (int q = 0; q < 8; ++q) t += red[q * 32 + s];
    atomicAdd(&est[n * S_D + s], t);  // global_atomic_add_f32
  }
}

// ---------------------------------------------------------------------------
// Kernel D: per-batch inclusive CDF of normalized weights.
//   grid = N, block = 1024, 16 elements/thread + Hillis-Steele scan.
// ---------------------------------------------------------------------------
__global__ __launch_bounds__(1024) void pf_cdf(const float* __restrict__ lw,
                                               const float* __restrict__ Z,
                                               float* __restrict__ cdf,
                                               float* __restrict__ tot) {
  __shared__ float red[1024];
  const int n = blockIdx.x, t = threadIdx.x;
  const size_t base = (size_t)n * M_P;
  const float z = Z[n];
  float loc[16];
  float s = 0.f;
#pragma unroll
  for (int i = 0; i < 16; ++i) {
    s += __expf(lw[base + t * 16 + i] - z);
    loc[i] = s;
  }
  red[t] = s;
  __syncthreads();
  for (int d = 1; d < 1024; d <<= 1) {
    const float v = (t >= d) ? red[t - d] : 0.f;
    __syncthreads();
    red[t] += v;
    __syncthreads();
  }
  const float off = (t > 0) ? red[t - 1] : 0.f;
#pragma unroll
  for (int i = 0; i < 16; ++i) cdf[base + t * 16 + i] = off + loc[i];
  if (t == 1023) tot[n] = red[1023];
}

// ---------------------------------------------------------------------------
// Kernel E: categorical resample via inverse-CDF + coalesced row gather.
//   grid = (M/8, N), block = 256 (8 waves, one sample per wave).
// ---------------------------------------------------------------------------
__device__ __forceinline__ unsigned pf_hash(unsigned x) {
  x ^= x >> 16; x *= 0x7feb352dU;
  x ^= x >> 15; x *= 0x846ca68bU;
  x ^= x >> 16;
  return x;
}

__global__ __launch_bounds__(256) void pf_resample(
    const float* __restrict__ pred, const float* __restrict__ cdf,
    const float* __restrict__ tot, float* __restrict__ out_states,
    float* __restrict__ out_lw) {
  const int n = blockIdx.y;
  const int g = threadIdx.x >> 5, lane = threadIdx.x & 31;
  const int j = blockIdx.x * 8 + g;
  const size_t base = (size_t)n * M_P;
  int idx = 0;
  if (lane == 0) {
    const unsigned h =
        pf_hash((unsigned)(n * M_P + j) * 2654435761u + 0x9e3779b9u);
    const float u = (float)(h >> 8) * (1.0f / 16777216.0f);
    const float target = u * tot[n];
    int lo = 0, hi = M_P - 1;
    while (lo < hi) {  // 14 steps
      const int mid = (lo + hi) >> 1;
      if (cdf[base + mid] < target) lo = mid + 1; else hi = mid;
    }
    idx = lo;
  }
  idx = __shfl(idx, 0);
  out_states[(base + j) * S_D + lane] = pred[(base + (size_t)idx) * S_D + lane];
  if (lane == 0) out_lw[base + j] = -__logf((float)M_P);
}

// ---------------------------------------------------------------------------
extern "C" void kernel_launch(void* const* d_in, const int* in_sizes, int n_in,
                              void* d_out, int out_size, void* d_ws,
                              size_t ws_size, hipStream_t stream) {
  const float* sp    = (const float*)d_in[0];  // states_prev   (N,M,S)
  const float* lwp   = (const float*)d_in[1];  // log_weights   (N,M)
  const float* obs   = (const float*)d_in[2];  // observations  (N,O)
  const float* ctrl  = (const float*)d_in[3];  // controls      (N,C)
  const float* Wd    = (const float*)d_in[4];  // W_dyn         (S,S)
  const float* Wc    = (const float*)d_in[5];  // W_ctrl        (C,S)
  const float* Wo    = (const float*)d_in[6];  // W_obs         (S,O)
  const float* noise = (const float*)d_in[7];  // noise         (N,M,S)

  float* out = (float*)d_out;
  float* out_est    = out;                                  // N*S
  float* out_states = out + N_B * S_D;                      // N*M*S
  float* out_lw     = out_states + (size_t)N_B * M_P * S_D; // N*M

  float* ws      = (float*)d_ws;
  float* ws_pred = ws;                                   // N*M*S
  float* ws_lw   = ws_pred + (size_t)N_B * M_P * S_D;    // N*M
  float* ws_Z    = ws_lw + (size_t)N_B * M_P;            // N
  float* ws_tot  = ws_Z + N_B;                           // N
  float* ws_cdf  = ws_tot + N_B;                         // N*M

  pf_predict_ll<<<dim3(M_P / 128, N_B), 256, 0, stream>>>(
      sp, lwp, obs, ctrl, Wd, Wc, Wo, noise, ws_pred, ws_lw);
  pf_lse<<<N_B, 256, 0, stream>>>(ws_lw, ws_Z);
  pf_zero<<<(N_B * S_D + 255) / 256, 256, 0, stream>>>(out_est, N_B * S_D);
  pf_estimate<<<dim3(8, N_B), 256, 0, stream>>>(ws_pred, ws_lw, ws_Z, out_est);
  pf_cdf<<<N_B, 1024, 0, stream>>>(ws_lw, ws_Z, ws_cdf, ws_tot);
  pf_resample<<<dim3(M_P / 8, N_B), 256, 0, stream>>>(ws_pred, ws_cdf, ws_tot,
                                                      out_states, out_lw);
}